// QuanvolutionFilter_65481071410307
// MI455X (gfx1250) — compile-verified
//
#include <hip/hip_runtime.h>
#include <hip/hip_bf16.h>
#include <stdint.h>

// Quanvolution filter, analytically reduced:
//   out[b, (r*14+c)*4 + i] = prod_{q=0..i} cos(x[b, 2r+dr_q, 2c+dc_q])
// (RZ phases cancel; CNOT chain -> parity bits -> prefix products of cos).
// Memory-bound: ~51 MB total traffic -> ~2.2 us at 23.3 TB/s HBM.
// Data path: async global->LDS (ASYNCcnt) staging of 4 images per workgroup,
// then per-patch compute from LDS with v_cos_f32.

namespace {
constexpr int IMG_W      = 28;
constexpr int IMG_FLOATS = 28 * 28;              // 784 floats per image
constexpr int PATCHES    = 14 * 14;              // 196 patches per image
constexpr int IMGS       = 4;                    // images staged per block
constexpr int BLOCK      = 256;                  // 8 waves (wave32)
constexpr int CHUNK4     = IMG_FLOATS / 4;       // 196 float4 chunks per image
}

__global__ __launch_bounds__(BLOCK)
void quanv_kernel(const float* __restrict__ x, float* __restrict__ out, int nimg)
{
    __shared__ __align__(16) float smem[IMGS * IMG_FLOATS];   // 12,544 B of 320 KB

    const int t = threadIdx.x;
    const long long imgBase = (long long)blockIdx.x * IMGS;
    const long long elemBase = imgBase * IMG_FLOATS;

    int imgsHere = nimg - (int)imgBase;
    if (imgsHere > IMGS) imgsHere = IMGS;

    // ---- Stage imgsHere images into LDS via async 128-bit global->LDS DMA ----
    const float* gsrc = x + elemBase;
    const int chunksHere = imgsHere * CHUNK4;
    for (int j = t; j < chunksHere; j += BLOCK) {
        // LDS byte address = low 32 bits of the generic (flat) shared pointer.
        uint32_t lds_addr = (uint32_t)(uintptr_t)(&smem[j * 4]);
        const float* gaddr = gsrc + j * 4;
        asm volatile("global_load_async_to_lds_b128 %0, %1, off"
                     :
                     : "v"(lds_addr), "v"(gaddr)
                     : "memory");
    }
    // Wait for this wave's async transfers, then make LDS visible block-wide.
    __builtin_amdgcn_s_wait_asynccnt(0);
    __syncthreads();

    // ---- Per-patch compute: prefix products of cos over the 2x2 patch ----
    float* gout = out + elemBase;                 // out stride per image is also 784
    const int patchesHere = imgsHere * PATCHES;
    for (int p = t; p < patchesHere; p += BLOCK) {
        const int img = p / PATCHES;
        const int pp  = p - img * PATCHES;        // r*14 + c
        const int r   = pp / 14;
        const int c   = pp - r * 14;
        const float* s = &smem[img * IMG_FLOATS + (2 * r) * IMG_W + 2 * c];

        const float z0 = __cosf(s[0]);
        const float z1 = z0 * __cosf(s[1]);
        const float z2 = z1 * __cosf(s[IMG_W]);
        const float z3 = z2 * __cosf(s[IMG_W + 1]);

        *reinterpret_cast<float4*>(gout + (long long)p * 4) =
            make_float4(z0, z1, z2, z3);
    }
}

extern "C" void kernel_launch(void* const* d_in, const int* in_sizes, int n_in,
                              void* d_out, int out_size, void* d_ws, size_t ws_size,
                              hipStream_t stream)
{
    const float* x = (const float*)d_in[0];
    // d_in[1] (weights) is mathematically irrelevant: the RZ phases cancel in
    // the Z-basis probabilities, so the reference output does not depend on it.
    float* out = (float*)d_out;

    const int nimg  = in_sizes[0] / IMG_FLOATS;          // B = 8192
    const int blocks = (nimg + IMGS - 1) / IMGS;         // 2048 workgroups
    quanv_kernel<<<blocks, BLOCK, 0, stream>>>(x, out, nimg);
}